// Attention_24455543783995
// MI455X (gfx1250) — compile-verified
//
#include <hip/hip_runtime.h>

#define B_     512
#define H_     32
#define D_     64
#define S_     64
#define START_ 32
#define STEPS_ 32
#define BT_    8      // batch tile per workgroup
#define KVS    68     // padded LDS row stride (floats) -> avoids 32-way bank conflicts

typedef __attribute__((ext_vector_type(2))) float v2f;
typedef __attribute__((ext_vector_type(4))) float v4f;
typedef __attribute__((ext_vector_type(8))) float v8f;
typedef __attribute__((ext_vector_type(4))) unsigned int v4u;
typedef __attribute__((ext_vector_type(4))) int v4i;
typedef __attribute__((ext_vector_type(8))) int v8i;

// LDS layout (float offsets)
#define OFF_K 0
#define OFF_V (BT_ * S_ * KVS)            // 34816
#define OFF_X (2 * BT_ * S_ * KVS)        // 69632
#define OFF_Q (OFF_X + 16 * 64)           // 70656
#define OFF_P (OFF_Q + 16 * 64)           // 71680
#define LDS_FLOATS (OFF_P + BT_ * 64)     // 72192 floats = 288768 bytes (<320KB)

// TDM descriptor group1 word0:
//   data_size=2 (4B)            -> bits[17:16] = 2      : 0x00020000
//   pad_enable=1                -> bit[20]              : 0x00100000
//   pad_interval=5 (64 DWORDs)  -> bits[24:22] = 5      : 0x01400000
//   pad_amount=3  (4 DWORDs)    -> bits[31:25] = 3      : 0x06000000
#define TDM_G1_W0 0x07520000

__global__ __launch_bounds__(256, 1)
void attn_decode_kernel(const float* __restrict__ x,
                        const float* __restrict__ k,
                        const float* __restrict__ v,
                        const float* __restrict__ wq,
                        const float* __restrict__ wk,
                        const float* __restrict__ wv,
                        const float* __restrict__ wo,
                        float* __restrict__ k_out,
                        float* __restrict__ v_out,
                        float* __restrict__ x_out)
{
    extern __shared__ float smem[];
    float* Kc = smem + OFF_K;   // [BT_][S_][KVS]
    float* Vc = smem + OFF_V;   // [BT_][S_][KVS]
    float* Xb = smem + OFF_X;   // [16][64]  rows 8..15 stay zero
    float* Qb = smem + OFF_Q;   // [16][64]  q / attn-out buffer
    float* Pb = smem + OFF_P;   // [BT_][64] softmax probs

    const int tid  = threadIdx.x;
    const int w    = tid >> 5;        // wave id 0..7
    const int lane = tid & 31;
    const int b0   = blockIdx.x * BT_;
    const int h    = blockIdx.y;

    // ---- TDM: async DMA of this wave's batch K/V 64x64 f32 tile into LDS,
    //      with hardware padding reproducing the 68-float LDS row stride ----
    {
        const int b = w;  // wave-uniform
        const size_t gbase = (((size_t)(b0 + b) * H_ + h) * S_) * D_;
        const unsigned long long gaK = (unsigned long long)(const void*)(k + gbase);
        const unsigned long long gaV = (unsigned long long)(const void*)(v + gbase);
        const unsigned ldsK = (unsigned)(unsigned long long)(const void*)(Kc + b * S_ * KVS);
        const unsigned ldsV = (unsigned)(unsigned long long)(const void*)(Vc + b * S_ * KVS);

        const v8i g1 = { (int)TDM_G1_W0,
                         64 << 16,        // tensor_dim0[15:0] in [31:16]
                         64 << 16,        // tensor_dim0 hi=0 | tensor_dim1[15:0]
                         64 << 16,        // tensor_dim1 hi=0 | tile_dim0
                         64,              // tile_dim1 | tile_dim2=0
                         64,              // tensor_dim0_stride lo
                         0, 0 };
        const v4i gz4 = { 0, 0, 0, 0 };
        const v8i gz8 = { 0, 0, 0, 0, 0, 0, 0, 0 };

        const v4u g0k = { 1u, ldsK, (unsigned)gaK,
                          (unsigned)((gaK >> 32) & 0x01FFFFFFu) | 0x80000000u }; // type=2
        const v4u g0v = { 1u, ldsV, (unsigned)gaV,
                          (unsigned)((gaV >> 32) & 0x01FFFFFFu) | 0x80000000u };

        __builtin_amdgcn_tensor_load_to_lds(g0k, g1, gz4, gz4, gz8, 0);
        __builtin_amdgcn_tensor_load_to_lds(g0v, g1, gz4, gz4, gz8, 0);
    }

    // ---- per-wave weight fragments (WMMA B operands, held in VGPRs all kernel) ----
    // waves 0-3: {Wq, Wv} column tiles; waves 4-7: {Wk, Wo} column tiles
    const float* W1 = (w < 4 ? wq : wk) + (size_t)h * D_ * D_;
    const float* W2 = (w < 4 ? wv : wo) + (size_t)h * D_ * D_;
    const int n   = w & 3;                 // N-tile (16 cols)
    const int col = n * 16 + (lane & 15);
    const int ko  = (lane >> 4) << 1;      // K sub-offset 0 or 2 (B/A fragment layout)

    __builtin_prefetch(W1 + (size_t)lane * 64, 0, 3);   // global_prefetch_b8
    __builtin_prefetch(W2 + (size_t)lane * 64, 0, 3);

    v2f wfA[16], wfB[16];
#pragma unroll
    for (int kk = 0; kk < 16; ++kk) {
        const int r = kk * 4 + ko;
        wfA[kk].x = W1[(r + 0) * D_ + col];
        wfA[kk].y = W1[(r + 1) * D_ + col];
        wfB[kk].x = W2[(r + 0) * D_ + col];
        wfB[kk].y = W2[(r + 1) * D_ + col];
    }

    // ---- load x into Xb rows 0..7, zero rows 8..15 (overlaps the TDM DMA) ----
    for (int idx = tid; idx < 16 * 64; idx += 256) {
        const int m = idx >> 6, d = idx & 63;
        float val = 0.0f;
        if (m < BT_) val = x[((size_t)(b0 + m) * H_ + h) * D_ + d];
        Xb[idx] = val;
    }

    __builtin_amdgcn_s_wait_tensorcnt(0);
    __syncthreads();

    // ---- copy K/V rows 0..31 from LDS to the outputs (rows 32..63 are
    //      produced and stored during the decode loop) ----
    for (int q4 = tid; q4 < BT_ * START_ * (D_ / 4); q4 += 256) {
        const int m   = q4 >> 9;
        const int rem = q4 & 511;
        const int s   = rem >> 4;
        const int d4  = (rem & 15) << 2;
        const size_t g = (((size_t)(b0 + m) * H_ + h) * S_ + s) * D_ + d4;
        const int l = (m * S_ + s) * KVS + d4;
        *(v4f*)(k_out + g) = *(const v4f*)(Kc + l);
        *(v4f*)(v_out + g) = *(const v4f*)(Vc + l);
    }

    // 16x64 tile (LDS) times held 64x16 weight tile -> 16x16 f32 accumulator
    auto mma64 = [&](const float* src, const v2f* wf) -> v8f {
        v8f acc = {};
        const int m = lane & 15;
#pragma unroll
        for (int kk = 0; kk < 16; ++kk) {
            v2f a;
            a.x = src[m * 64 + kk * 4 + ko + 0];
            a.y = src[m * 64 + kk * 4 + ko + 1];
            acc = __builtin_amdgcn_wmma_f32_16x16x4_f32(
                false, a, false, wf[kk], (short)0, acc, false, false);
        }
        return acc;
    };

    for (int i = 0; i < STEPS_; ++i) {
        const int srow = START_ + i;

        // ---- Phase 1: Q (waves 0-3) | K_new (waves 4-7) ----
        {
            v8f acc = mma64(Xb, wfA);
            if (w < 4) {
#pragma unroll
                for (int r = 0; r < 8; ++r) {
                    const int m = r + ((lane >> 4) << 3);
                    Qb[m * 64 + col] = acc[r];      // rows 8..15 write zeros
                }
            } else if (lane < 16) {                 // only rows 0..7 are real batches
#pragma unroll
                for (int r = 0; r < 8; ++r) {
                    Kc[(r * S_ + srow) * KVS + col] = acc[r];
                    k_out[(((size_t)(b0 + r) * H_ + h) * S_ + srow) * D_ + col] = acc[r];
                }
            }
        }
        __syncthreads();

        // ---- Phase 2: V_new (waves 0-3) ----
        if (w < 4) {
            v8f acc = mma64(Xb, wfB);
            if (lane < 16) {
#pragma unroll
                for (int r = 0; r < 8; ++r) {
                    Vc[(r * S_ + srow) * KVS + col] = acc[r];
                    v_out[(((size_t)(b0 + r) * H_ + h) * S_ + srow) * D_ + col] = acc[r];
                }
            }
        }
        __syncthreads();

        // ---- Phase 3: attention, wave w handles batch b=w ----
        {
            const int b = w;
            const float* Kr = Kc + b * S_ * KVS;
            const float* Vr = Vc + b * S_ * KVS;
            const float* qv = Qb + b * 64;
            const int s0 = lane, s1 = lane + 32;
            float sc0 = 0.f, sc1 = 0.f;
#pragma unroll
            for (int d = 0; d < 64; d += 4) {
                const v4f qd = *(const v4f*)(qv + d);               // LDS broadcast
                const v4f k0 = *(const v4f*)(Kr + s0 * KVS + d);
                const v4f k1 = *(const v4f*)(Kr + s1 * KVS + d);
                sc0 += qd.x * k0.x + qd.y * k0.y + qd.z * k0.z + qd.w * k0.w;
                sc1 += qd.x * k1.x + qd.y * k1.y + qd.z * k1.z + qd.w * k1.w;
            }
            sc0 *= 0.125f; sc1 *= 0.125f;
            float mx = fmaxf(sc0, sc1);
#pragma unroll
            for (int off = 16; off > 0; off >>= 1)
                mx = fmaxf(mx, __shfl_xor(mx, off, 32));
            const float e0 = __expf(sc0 - mx);
            const float e1 = __expf(sc1 - mx);
            float sum = e0 + e1;
#pragma unroll
            for (int off = 16; off > 0; off >>= 1)
                sum += __shfl_xor(sum, off, 32);
            const float inv = 1.0f / sum;
            Pb[b * 64 + s0] = e0 * inv;
            Pb[b * 64 + s1] = e1 * inv;

            float o0 = 0.f, o1 = 0.f;
#pragma unroll 4
            for (int s = 0; s < 64; ++s) {
                const float p = Pb[b * 64 + s];                     // LDS broadcast
                o0 += p * Vr[s * KVS + lane];                       // conflict-free
                o1 += p * Vr[s * KVS + lane + 32];
            }
            Qb[b * 64 + lane]      = o0;   // attn-out replaces q; rows 8..15 stay 0
            Qb[b * 64 + lane + 32] = o1;
        }
        __syncthreads();

        // ---- Phase 4: output projection (waves 4-7): Xb = AttnOut @ Wo ----
        if (w >= 4) {
            v8f acc = mma64(Qb, wfB);
#pragma unroll
            for (int r = 0; r < 8; ++r) {
                const int m = r + ((lane >> 4) << 3);
                Xb[m * 64 + col] = acc[r];          // rows 8..15 rewritten as zeros
            }
        }
        __syncthreads();
    }

    // ---- epilogue: final x ----
    for (int idx = tid; idx < BT_ * 64; idx += 256) {
        const int m = idx >> 6, d = idx & 63;
        x_out[((size_t)(b0 + m) * H_ + h) * D_ + d] = Xb[idx];
    }
}

extern "C" void kernel_launch(void* const* d_in, const int* in_sizes, int n_in,
                              void* d_out, int out_size, void* d_ws, size_t ws_size,
                              hipStream_t stream) {
    (void)in_sizes; (void)n_in; (void)out_size; (void)d_ws; (void)ws_size;
    const float* x  = (const float*)d_in[0];
    const float* k  = (const float*)d_in[1];
    const float* v  = (const float*)d_in[2];
    const float* wq = (const float*)d_in[3];
    const float* wk = (const float*)d_in[4];
    const float* wv = (const float*)d_in[5];
    const float* wo = (const float*)d_in[6];

    float* k_out = (float*)d_out;
    float* v_out = k_out + (size_t)B_ * H_ * S_ * D_;
    float* x_out = v_out + (size_t)B_ * H_ * S_ * D_;

    dim3 grid(B_ / BT_, H_);   // 64 x 32 workgroups
    dim3 block(256);           // 8 waves (wave32)
    const size_t lds_bytes = (size_t)LDS_FLOATS * sizeof(float);
    attn_decode_kernel<<<grid, block, lds_bytes, stream>>>(
        x, k, v, wq, wk, wv, wo, k_out, v_out, x_out);
}